// DenseDet_36764920053807
// MI455X (gfx1250) — compile-verified
//
#include <hip/hip_runtime.h>
#include <math.h>

// DenseDet post-processing for MI455X (gfx1250, wave32).
// B=4, levels: 48x48 (s=8), 24x24 (s=16), 12x12 (s=32) -> 3024 points/image.
#define NCLS     80
#define IMGF     384.0f
#define NTOT     3024
#define N0       2304
#define N01      2880
#define BATCH    4
#define CONF_THRF 0.05f
#define NMS_IOUF  0.6f
#define MAXDET   100

__device__ __forceinline__ float sigf(float x) { return 1.0f / (1.0f + expf(-x)); }

__device__ __forceinline__ unsigned lds_addr_of(const void* p) {
    // generic -> AS3 addrspacecast yields the 32-bit LDS byte offset
    __attribute__((address_space(3))) const void* lp =
        (__attribute__((address_space(3))) const void*)p;
    return (unsigned)(unsigned long long)lp;
}

__device__ __forceinline__ void async_wait0() {
#if __has_builtin(__builtin_amdgcn_s_wait_asynccnt)
    __builtin_amdgcn_s_wait_asynccnt(0);
#else
    asm volatile("s_wait_asynccnt 0" ::: "memory");
#endif
}

// ---------------------------------------------------------------------------
// Kernel 1: per-point decode (sigmoid class max/argmax, quality, box decode)
// ---------------------------------------------------------------------------
__global__ void decode_kernel(const float* __restrict__ cls0, const float* __restrict__ box0, const float* __restrict__ q0,
                              const float* __restrict__ cls1, const float* __restrict__ box1, const float* __restrict__ q1,
                              const float* __restrict__ cls2, const float* __restrict__ box2, const float* __restrict__ q2,
                              float* __restrict__ dbx, float* __restrict__ dsc, int* __restrict__ dlb)
{
    int t = blockIdx.x * blockDim.x + threadIdx.x;
    if (t >= BATCH * NTOT) return;
    int b = t / NTOT;
    int p = t - b * NTOT;

    const float *cls, *box, *qm;
    int W, st, base;
    if (p < N0)       { cls = cls0; box = box0; qm = q0; W = 48; st = 8;  base = 0;   }
    else if (p < N01) { cls = cls1; box = box1; qm = q1; W = 24; st = 16; base = N0;  }
    else              { cls = cls2; box = box2; qm = q2; W = 12; st = 32; base = N01; }

    int idx   = p - base;
    int plane = W * W;
    int y = idx / W;
    int x = idx - y * W;

    const float* cb = cls + (size_t)b * NCLS * plane + idx;
    const float* bb = box + (size_t)b * 4 * plane + idx;
    __builtin_prefetch(bb, 0, 0);   // gfx1250: global_prefetch_b8

    // sigmoid is monotone: max/argmax on logits == max/argmax on sigmoid
    float best = cb[0];
    int lab = 0;
    for (int c = 1; c < NCLS; ++c) {
        float v = cb[(size_t)c * plane];
        if (v > best) { best = v; lab = c; }
    }
    float raw   = sigf(best);
    float qual  = sigf(qm[(size_t)b * plane + idx]);
    float score = fminf(fmaxf(raw * qual, 0.0f), 1.0f);

    float fst = (float)st;
    float px  = ((float)x + 0.5f) * fst;
    float py  = ((float)y + 0.5f) * fst;
    float l  = bb[0] * fst;
    float tt = bb[(size_t)plane] * fst;
    float r  = bb[2 * (size_t)plane] * fst;
    float d  = bb[3 * (size_t)plane] * fst;
    float x1 = fminf(fmaxf(px - l,  0.0f), IMGF);
    float y1 = fminf(fmaxf(py - tt, 0.0f), IMGF);
    float x2 = fminf(fmaxf(px + r,  0.0f), IMGF);
    float y2 = fminf(fmaxf(py + d,  0.0f), IMGF);

    size_t o = (size_t)b * NTOT + p;
    dbx[o * 4 + 0] = x1; dbx[o * 4 + 1] = y1;
    dbx[o * 4 + 2] = x2; dbx[o * 4 + 3] = y2;
    dsc[o] = score;
    dlb[o] = lab;
}

// ---------------------------------------------------------------------------
// Kernel 2: stable descending rank sort (== jnp.argsort(-s), index tie-break)
// One block per image; scores async-DMA'd into LDS, O(N^2) rank, collision-
// free scatter (ranks are a permutation because ties break by index).
// ---------------------------------------------------------------------------
__global__ __launch_bounds__(1024)
void rank_kernel(const float* __restrict__ dbx, const float* __restrict__ dsc, const int* __restrict__ dlb,
                 float* __restrict__ sbx, float* __restrict__ ssc, int* __restrict__ slb)
{
    __shared__ float key[NTOT];
    const int tid = threadIdx.x;
    const int b   = blockIdx.x;
    const size_t g = (size_t)b * NTOT;

    // Async DMA raw scores into LDS (gfx1250 global_load_async_to_lds_b32)
    for (int i = tid; i < NTOT; i += blockDim.x) {
        unsigned long long ga = (unsigned long long)(size_t)(const void*)(dsc + g + i);
        unsigned la = lds_addr_of(&key[i]);
        asm volatile("global_load_async_to_lds_b32 %0, %1, off"
                     :: "v"(la), "v"(ga) : "memory");
    }
    async_wait0();
    __syncthreads();
    // Threshold in place: key = valid ? score : -1   (reference semantics)
    for (int i = tid; i < NTOT; i += blockDim.x) {
        float s = key[i];
        key[i] = (s > CONF_THRF) ? s : -1.0f;
    }
    __syncthreads();

    for (int i = tid; i < NTOT; i += blockDim.x) {
        float ki = key[i];
        int r = 0;
        for (int j = 0; j < NTOT; ++j) {
            float kj = key[j];                  // LDS broadcast read
            r += (int)((kj > ki) || (kj == ki && j < i));
        }
        size_t src = g + i;
        size_t dst = g + (size_t)r;
        sbx[dst * 4 + 0] = dbx[src * 4 + 0];
        sbx[dst * 4 + 1] = dbx[src * 4 + 1];
        sbx[dst * 4 + 2] = dbx[src * 4 + 2];
        sbx[dst * 4 + 3] = dbx[src * 4 + 3];
        ssc[dst] = ki;
        slb[dst] = dlb[src];
    }
}

// ---------------------------------------------------------------------------
// Kernel 3: exact greedy class-aware NMS + top-100 + cxcywh output.
// Boxes staged into LDS via CDNA5 async global->LDS DMA (ASYNCcnt tracked).
// Class-offset trick == same-label check (offsets >= 385 kill cross-class IoU).
// Early exit: once MAXDET boxes are kept, later keep-status cannot affect the
// output (output = first MAXDET kept, suppression only tests kept j < i).
// ---------------------------------------------------------------------------
__global__ __launch_bounds__(1024)
void nms_kernel(const float4* __restrict__ sbx, const float* __restrict__ ssc,
                const int* __restrict__ slb, float* __restrict__ out)
{
    __shared__ float4        boxs[NTOT];   // 48384 B
    __shared__ unsigned char labs[NTOT];   //  3024 B
    __shared__ unsigned char state[NTOT];  //  3024 B (valid -> kept)
    __shared__ int           ssup;
    __shared__ int           skept;

    const int tid  = threadIdx.x;
    const int b    = blockIdx.x;
    const size_t g = (size_t)b * NTOT;

    if (tid == 0) { ssup = 0; skept = 0; }
    for (int i = tid; i < NTOT; i += blockDim.x) {
        // Async DMA: global float4 -> LDS (gfx1250 global_load_async_to_lds_b128)
        unsigned long long ga = (unsigned long long)(size_t)(const void*)(sbx + g + i);
        unsigned la = lds_addr_of(&boxs[i]);
        asm volatile("global_load_async_to_lds_b128 %0, %1, off"
                     :: "v"(la), "v"(ga) : "memory");

        float s  = ssc[g + i];
        state[i] = (s > CONF_THRF) ? (unsigned char)1 : (unsigned char)0;
        labs[i]  = (unsigned char)slb[g + i];
    }
    async_wait0();
    __syncthreads();

    // Sequential greedy scan: keep[i] = valid[i] && !any(keep[j<i] && IoU > thr)
    for (int i = 0; i < NTOT; ++i) {
        if (state[i]) {
            float4 bi = boxs[i];
            int    li = labs[i];
            float  ai = (bi.z - bi.x) * (bi.w - bi.y);
            bool sup = false;
            for (int j = tid; j < i; j += blockDim.x) {
                if (state[j] && labs[j] == (unsigned char)li) {
                    float4 bj  = boxs[j];
                    float xx1 = fmaxf(bi.x, bj.x);
                    float yy1 = fmaxf(bi.y, bj.y);
                    float xx2 = fminf(bi.z, bj.z);
                    float yy2 = fminf(bi.w, bj.w);
                    float inter = fmaxf(xx2 - xx1, 0.0f) * fmaxf(yy2 - yy1, 0.0f);
                    float uni   = ai + (bj.z - bj.x) * (bj.w - bj.y) - inter;
                    if (inter / fmaxf(uni, 1e-9f) > NMS_IOUF) { sup = true; break; }
                }
            }
            if (sup) atomicOr(&ssup, 1);
        }
        __syncthreads();
        if (tid == 0) {
            if (ssup)            state[i] = 0;   // suppressed
            else if (state[i])   ++skept;        // kept
            ssup = 0;
        }
        __syncthreads();
        if (skept >= MAXDET) break;              // uniform: skept stable post-barrier
    }

    // Emit first MAXDET kept (already in score order); zero-fill the rest.
    if (tid == 0) {
        float* ob = out + (size_t)b * MAXDET * 4;
        float* os = out + (size_t)BATCH * MAXDET * 4 + (size_t)b * MAXDET;
        float* ol = out + (size_t)BATCH * MAXDET * 5 + (size_t)b * MAXDET;
        int cnt = 0;
        for (int i = 0; i < NTOT && cnt < MAXDET; ++i) {
            if (state[i]) {
                float4 bx = boxs[i];
                ob[cnt * 4 + 0] = (bx.x + bx.z) * 0.5f / IMGF;
                ob[cnt * 4 + 1] = (bx.y + bx.w) * 0.5f / IMGF;
                ob[cnt * 4 + 2] = (bx.z - bx.x) / IMGF;
                ob[cnt * 4 + 3] = (bx.w - bx.y) / IMGF;
                os[cnt] = ssc[g + i];
                ol[cnt] = (float)labs[i];
                ++cnt;
            }
        }
        for (; cnt < MAXDET; ++cnt) {
            ob[cnt * 4 + 0] = 0.0f; ob[cnt * 4 + 1] = 0.0f;
            ob[cnt * 4 + 2] = 0.0f; ob[cnt * 4 + 3] = 0.0f;
            os[cnt] = 0.0f; ol[cnt] = 0.0f;
        }
    }
}

// ---------------------------------------------------------------------------
extern "C" void kernel_launch(void* const* d_in, const int* in_sizes, int n_in,
                              void* d_out, int out_size, void* d_ws, size_t ws_size,
                              hipStream_t stream) {
    (void)in_sizes; (void)n_in; (void)out_size; (void)ws_size;
    // setup_inputs() dict order: cls0, box0, q0, cls1, box1, q1, cls2, box2, q2
    const float* cls0 = (const float*)d_in[0];
    const float* box0 = (const float*)d_in[1];
    const float* q0   = (const float*)d_in[2];
    const float* cls1 = (const float*)d_in[3];
    const float* box1 = (const float*)d_in[4];
    const float* q1   = (const float*)d_in[5];
    const float* cls2 = (const float*)d_in[6];
    const float* box2 = (const float*)d_in[7];
    const float* q2   = (const float*)d_in[8];
    float* out = (float*)d_out;

    // Workspace layout (floats): dec boxes/scores/labels, sorted boxes/scores/labels
    float* dbx = (float*)d_ws;                       // B*NTOT*4
    float* dsc = dbx + (size_t)BATCH * NTOT * 4;     // B*NTOT
    int*   dlb = (int*)(dsc + (size_t)BATCH * NTOT); // B*NTOT
    float* sbx = (float*)(dlb + (size_t)BATCH * NTOT);       // B*NTOT*4 (16B-aligned: 290304 % 16 == 0)
    float* ssc = sbx + (size_t)BATCH * NTOT * 4;             // B*NTOT
    int*   slb = (int*)(ssc + (size_t)BATCH * NTOT);         // B*NTOT

    int total = BATCH * NTOT;
    decode_kernel<<<(total + 255) / 256, 256, 0, stream>>>(cls0, box0, q0, cls1, box1, q1,
                                                           cls2, box2, q2, dbx, dsc, dlb);
    rank_kernel<<<BATCH, 1024, 0, stream>>>(dbx, dsc, dlb, sbx, ssc, slb);
    nms_kernel<<<BATCH, 1024, 0, stream>>>((const float4*)sbx, ssc, slb, out);
}